// FewShotNoEmbeddingPredictor_29609504539515
// MI455X (gfx1250) — compile-verified
//
#include <hip/hip_runtime.h>
#include <cmath>

typedef __attribute__((ext_vector_type(2))) float v2f;
typedef __attribute__((ext_vector_type(8))) float v8f;

constexpr int WAVES = 8;
constexpr int BLOCK = 256;
constexpr int ITEMS_PER_WAVE = 16;
constexpr int ITEMS_PER_BLOCK = WAVES * ITEMS_PER_WAVE;  // 128
constexpr int AST = 116;   // row-major activation LDS stride (floats), bank-conflict-free for A reads
constexpr int WST = 112;   // weight logical row width (floats) = 7 tiles of 16
constexpr int WPS = 2 * WST;  // 224: paired-K row stride (two K-rows interleaved per column)

// ---- LDS layout (floats) ----
// Weights are stored K-pair-interleaved: element (k,n) -> [(k>>1)*WPS + n*2 + (k&1)]
// so a B fragment (k,k+1 @ col n) is one aligned ds_load_b64.
constexpr int OFF_GW1 = 0;                        // gc1: 8   x 112 (4 pair-rows)
constexpr int OFF_GW2 = OFF_GW1 + 8   * WST;      // gc2: 100 x 112 (50 pair-rows)
constexpr int OFF_GW3 = OFF_GW2 + 100 * WST;      // gc3
constexpr int OFF_GW4 = OFF_GW3 + 100 * WST;      // gc4
constexpr int OFF_GCB = OFF_GW4 + 100 * WST;      // 4 x 112 GCN biases
constexpr int OFF_SB  = OFF_GCB + 4 * WST;        // stage bias (112)
constexpr int OFF_ACT = OFF_SB + WST;             // per-wave activation buffers
constexpr int ACTSZ   = 16 * AST;                 // 1856 floats (tbuf uses 8*224=1792 of it)
constexpr int LDS_FLOATS = OFF_ACT + 3 * WAVES * ACTSZ;   // 79,600 floats
constexpr size_t LDS_BYTES = (size_t)LDS_FLOATS * sizeof(float);  // 318,400 B <= 320KB

// ---- WMMA fp32 16x16x4 wrapper ----
__device__ __forceinline__ v8f wmma4(v2f a, v2f b, v8f c) {
  return __builtin_amdgcn_wmma_f32_16x16x4_f32(
      /*neg_a=*/false, a, /*neg_b=*/false, b,
      /*c_mod=*/(short)0, c, /*reuse_a=*/false, /*reuse_b=*/false);
}

__device__ __forceinline__ v8f bias_tile(const float* b, int n0, int lane) {
  float bv = b[n0 + (lane & 15)];
  v8f c;
#pragma unroll
  for (int r = 0; r < 8; ++r) c[r] = bv;
  return c;
}

__device__ __forceinline__ v8f zero_tile() {
  v8f c;
#pragma unroll
  for (int r = 0; r < 8; ++r) c[r] = 0.0f;
  return c;
}

// C-tile (16x16) store to a row-major activation buffer (stride AST).
__device__ __forceinline__ void store_tile(float* act, int n0, v8f c, int lane, bool relu) {
  const int n = lane & 15;
  const int hi = lane >> 4;
#pragma unroll
  for (int r = 0; r < 8; ++r) {
    float v = c[r];
    if (relu) v = fmaxf(v, 0.0f);
    act[(r + 8 * hi) * AST + n0 + n] = v;
  }
}

// C-tile store to the K-pair-interleaved t buffer: rows (2j,2j+1) are an aligned b64 pair.
__device__ __forceinline__ void store_tile_pairs(float* tp, int n0, v8f c, int lane) {
  const int n = lane & 15;
  const int hi = lane >> 4;
#pragma unroll
  for (int j = 0; j < 4; ++j) {
    v2f v;
    v.x = c[2 * j];
    v.y = c[2 * j + 1];
    *(v2f*)(tp + (4 * hi + j) * WPS + (n0 + n) * 2) = v;
  }
}

// Masked store: keep only columns < 20, write at dstoff + col (row-major buffer).
__device__ __forceinline__ void store_mask20(float* act, int n0, int dstoff, v8f c, int lane, bool relu) {
  const int n = lane & 15;
  const int hi = lane >> 4;
  const int col = n0 + n;
  if (col < 20) {
#pragma unroll
    for (int r = 0; r < 8; ++r) {
      float v = c[r];
      if (relu) v = fmaxf(v, 0.0f);
      act[(r + 8 * hi) * AST + dstoff + col] = v;
    }
  }
}

// ---- n-tile-resident GEMM stages: k outer loop, NT independent accumulators ----
// A fragment: one ds_load_b64 per k-step; B fragment: one ds_load_b64 (paired layout).

// A from row-major LDS activations, B from paired LDS weights.
template <int NT, bool RELU, bool POUT>
__device__ __forceinline__ void stage_lds(const float* __restrict__ in, const float* __restrict__ w,
                                          const float* bias, float* __restrict__ outb,
                                          int ksteps, int lane) {
  const int n = lane & 15;             // A-row == B-col lane index
  const int k2 = (lane >> 4) * 2;      // 0 or 2
  v8f c[NT];
#pragma unroll
  for (int t = 0; t < NT; ++t) c[t] = bias ? bias_tile(bias, t * 16, lane) : zero_tile();
  for (int kk = 0; kk < ksteps; ++kk) {
    const int k = 4 * kk + k2;
    v2f a = *(const v2f*)(in + n * AST + k);
    const float* wp = w + (k >> 1) * WPS;
#pragma unroll
    for (int t = 0; t < NT; ++t) {
      v2f b = *(const v2f*)(wp + (t * 16 + n) * 2);
      c[t] = wmma4(a, b, c[t]);
    }
  }
#pragma unroll
  for (int t = 0; t < NT; ++t) {
    if (POUT) store_tile_pairs(outb, t * 16, c[t], lane);
    else      store_tile(outb, t * 16, c[t], lane, RELU);
  }
}

// A from a per-lane global row pointer (feat / hw_emb), B from paired LDS weights.
template <int NT, bool RELU, bool POUT>
__device__ __forceinline__ void stage_glb(const float* __restrict__ grow, const float* __restrict__ w,
                                          const float* bias, float* __restrict__ outb,
                                          int ksteps, int lane) {
  const int n = lane & 15;
  const int k2 = (lane >> 4) * 2;
  v8f c[NT];
#pragma unroll
  for (int t = 0; t < NT; ++t) c[t] = bias ? bias_tile(bias, t * 16, lane) : zero_tile();
  for (int kk = 0; kk < ksteps; ++kk) {
    const int k = 4 * kk + k2;
    v2f a = *(const v2f*)(grow + k);
    const float* wp = w + (k >> 1) * WPS;
#pragma unroll
    for (int t = 0; t < NT; ++t) {
      v2f b = *(const v2f*)(wp + (t * 16 + n) * 2);
      c[t] = wmma4(a, b, c[t]);
    }
  }
#pragma unroll
  for (int t = 0; t < NT; ++t) {
    if (POUT) store_tile_pairs(outb, t * 16, c[t], lane);
    else      store_tile(outb, t * 16, c[t], lane, RELU);
  }
}

// N=20 stage (2 n-tiles), masked store into row-major buffer at dstoff.
__device__ __forceinline__ void stage_mask20(const float* __restrict__ in, const float* __restrict__ w,
                                             const float* bias, float* __restrict__ outb,
                                             int dstoff, int ksteps, int lane, bool relu) {
  const int n = lane & 15;
  const int k2 = (lane >> 4) * 2;
  v8f c0 = bias_tile(bias, 0, lane);
  v8f c1 = bias_tile(bias, 16, lane);
  for (int kk = 0; kk < ksteps; ++kk) {
    const int k = 4 * kk + k2;
    v2f a = *(const v2f*)(in + n * AST + k);
    const float* wp = w + (k >> 1) * WPS;
    v2f b0 = *(const v2f*)(wp + n * 2);
    v2f b1 = *(const v2f*)(wp + (16 + n) * 2);
    c0 = wmma4(a, b0, c0);
    c1 = wmma4(a, b1, c1);
  }
  store_mask20(outb, 0, dstoff, c0, lane, relu);
  store_mask20(outb, 16, dstoff, c1, lane, relu);
}

// h_new = relu( blockdiag(adj0,adj1) @ t + bias ); t is K-pair-interleaved in LDS.
__device__ __forceinline__ void adj_stage(const v2f* af, const float* __restrict__ t,
                                          const float* bias, float* __restrict__ h, int lane) {
  const int n = lane & 15;
  const int hi = lane >> 4;
  v8f c[7];
#pragma unroll
  for (int tt = 0; tt < 7; ++tt) c[tt] = bias_tile(bias, tt * 16, lane);
#pragma unroll
  for (int kk = 0; kk < 4; ++kk) {
    const float* tp = t + (2 * kk + hi) * WPS;   // (k>>1) with k = 4*kk + 2*hi
#pragma unroll
    for (int tt = 0; tt < 7; ++tt) {
      v2f b = *(const v2f*)(tp + (tt * 16 + n) * 2);
      c[tt] = wmma4(af[kk], b, c[tt]);
    }
  }
#pragma unroll
  for (int tt = 0; tt < 7; ++tt) store_tile(h, tt * 16, c[tt], lane, true);
}

// Cooperative weight load: src (KxN row-major) -> paired LDS layout, zero-padded cols.
__device__ __forceinline__ void load_w_lds(float* dst, const float* src, int K, int N) {
  const int total = K * WST;
  for (int i = threadIdx.x; i < total; i += BLOCK) {
    const int r = i / WST;
    const int c = i - r * WST;
    dst[(r >> 1) * WPS + c * 2 + (r & 1)] = (c < N) ? src[r * N + c] : 0.0f;
  }
}

__device__ __forceinline__ void load_b_lds(float* dst, const float* src, int N) {
  for (int i = threadIdx.x; i < WST; i += BLOCK) dst[i] = (i < N) ? src[i] : 0.0f;
}

__global__ __launch_bounds__(BLOCK)
void predictor_kernel(const float* __restrict__ feat, const float* __restrict__ adj,
                      const float* __restrict__ hw_emb,
                      const float* gc1_w, const float* gc1_b, const float* gc2_w, const float* gc2_b,
                      const float* gc3_w, const float* gc3_b, const float* gc4_w, const float* gc4_b,
                      const float* fc3_w, const float* fc3_b, const float* fc4_w, const float* fc4_b,
                      const float* fc5_w, const float* fc5_b,
                      const float* f0_w, const float* f0_b, const float* f1_w, const float* f1_b,
                      const float* f2_w, const float* f2_b, const float* f3_w, const float* f3_b,
                      const float* f4_w, const float* f4_b,
                      const float* e0_w, const float* e0_b, const float* e1_w, const float* e1_b,
                      float* __restrict__ out) {
  extern __shared__ float lds[];
  const int tid = threadIdx.x;
  const int wave = tid >> 5;
  const int lane = tid & 31;
  const int n = lane & 15;
  const int hi = lane >> 4;
  const int k2 = hi * 2;
  const int itembase = blockIdx.x * ITEMS_PER_BLOCK + wave * ITEMS_PER_WAVE;

  float* hbuf = lds + OFF_ACT + (0 * WAVES + wave) * ACTSZ;  // row-major (A source)
  float* tbuf = lds + OFF_ACT + (1 * WAVES + wave) * ACTSZ;  // K-pair-interleaved (B source)
  float* gbuf = lds + OFF_ACT + (2 * WAVES + wave) * ACTSZ;  // row-major (A source)

  // ---- Stage all GCN weights/biases into LDS ----
  load_w_lds(lds + OFF_GW1, gc1_w, 8, 100);
  load_w_lds(lds + OFF_GW2, gc2_w, 100, 100);
  load_w_lds(lds + OFF_GW3, gc3_w, 100, 100);
  load_w_lds(lds + OFF_GW4, gc4_w, 100, 100);
  load_b_lds(lds + OFF_GCB + 0 * WST, gc1_b, 100);
  load_b_lds(lds + OFF_GCB + 1 * WST, gc2_b, 100);
  load_b_lds(lds + OFF_GCB + 2 * WST, gc3_b, 100);
  load_b_lds(lds + OFF_GCB + 3 * WST, gc4_b, 100);
  __syncthreads();

  // ---- Phase A: GCN, 2 items per pass (16-row block-diagonal WMMA) ----
  const int rb = n >> 3;   // which item of the pair this row belongs to (0/1)
  const int r8 = n & 7;    // node index within the item
  for (int p = 0; p < 8; ++p) {
    const int item = itembase + 2 * p + rb;
    const float* featrow = feat + ((size_t)item * 8 + r8) * 8;
    const float* adjrow  = adj  + ((size_t)item * 8 + r8) * 8;
    if (p < 7) __builtin_prefetch(adj + (size_t)(item + 2) * 64, 0, 1);

    // Preload block-diagonal adj A-fragments (16x16, 4 k-steps) into registers.
    v2f af[4];
#pragma unroll
    for (int kk = 0; kk < 4; ++kk) {
      const int kloc = (kk & 1) * 4 + k2;       // column within the item's 8x8 block
      v2f a = *(const v2f*)(adjrow + kloc);
      if (rb != (kk >> 1)) { a.x = 0.0f; a.y = 0.0f; }  // off-diagonal block = 0
      af[kk] = a;
    }

    // Layer 1: t = feat(16x8) @ gc1_w ; h = relu(adj@t + b1)
    stage_glb<7, false, true>(featrow, lds + OFF_GW1, nullptr, tbuf, 2, lane);
    adj_stage(af, tbuf, lds + OFF_GCB + 0 * WST, hbuf, lane);

    // Layers 2..4: t = h @ gcL_w ; h = relu(adj@t + bL)
    const int gwoff[3] = {OFF_GW2, OFF_GW3, OFF_GW4};
#pragma unroll
    for (int L = 0; L < 3; ++L) {
      stage_lds<7, false, true>(hbuf, lds + gwoff[L], nullptr, tbuf, 25, lane);
      adj_stage(af, tbuf, lds + OFF_GCB + (L + 1) * WST, hbuf, lane);
    }

    // Extract node 7 (last node) of each item -> gbuf rows 2p / 2p+1
    for (int nt = 0; nt < 7; ++nt)
      gbuf[(2 * p + hi) * AST + nt * 16 + n] = hbuf[(7 + 8 * hi) * AST + nt * 16 + n];
  }
  __syncthreads();

  float* wbuf  = lds;                // reuse GCN weight region for stage weights
  float* sbias = lds + OFF_SB;

  // ---- Phase B: vector stages, M=16 items per wave (row-major ping-pong: gbuf<->hbuf) ----
  // fc3: g (gbuf) -> hbuf (relu)
  load_w_lds(wbuf, fc3_w, 100, 100); load_b_lds(sbias, fc3_b, 100); __syncthreads();
  stage_lds<7, true, false>(gbuf, wbuf, sbias, hbuf, 25, lane);
  __syncthreads();
  // fc4: hbuf -> gbuf (relu)
  load_w_lds(wbuf, fc4_w, 100, 100); load_b_lds(sbias, fc4_b, 100); __syncthreads();
  stage_lds<7, true, false>(hbuf, wbuf, sbias, gbuf, 25, lane);
  __syncthreads();
  // fc5: gbuf -> hbuf[:,0:20] (no relu)
  load_w_lds(wbuf, fc5_w, 100, 20); load_b_lds(sbias, fc5_b, 20); __syncthreads();
  stage_mask20(gbuf, wbuf, sbias, hbuf, 0, 25, lane, false);
  __syncthreads();
  // e0: hw_emb (global A-frags, K=20) -> gbuf (relu)
  load_w_lds(wbuf, e0_w, 20, 100); load_b_lds(sbias, e0_b, 100); __syncthreads();
  stage_glb<7, true, false>(hw_emb + (size_t)(itembase + n) * 20, wbuf, sbias, gbuf, 5, lane);
  __syncthreads();
  // e1: gbuf -> hbuf[:,20:40] (relu)  => hbuf[:,0:40] = concat(g, hw)
  load_w_lds(wbuf, e1_w, 100, 20); load_b_lds(sbias, e1_b, 20); __syncthreads();
  stage_mask20(gbuf, wbuf, sbias, hbuf, 20, 25, lane, true);
  __syncthreads();
  // f0: concat in hbuf (K=40) -> gbuf (relu)
  load_w_lds(wbuf, f0_w, 40, 100); load_b_lds(sbias, f0_b, 100); __syncthreads();
  stage_lds<7, true, false>(hbuf, wbuf, sbias, gbuf, 10, lane);
  __syncthreads();
  // f1: gbuf -> hbuf
  load_w_lds(wbuf, f1_w, 100, 100); load_b_lds(sbias, f1_b, 100); __syncthreads();
  stage_lds<7, true, false>(gbuf, wbuf, sbias, hbuf, 25, lane);
  __syncthreads();
  // f2: hbuf -> gbuf
  load_w_lds(wbuf, f2_w, 100, 100); load_b_lds(sbias, f2_b, 100); __syncthreads();
  stage_lds<7, true, false>(hbuf, wbuf, sbias, gbuf, 25, lane);
  __syncthreads();
  // f3: gbuf -> hbuf
  load_w_lds(wbuf, f3_w, 100, 100); load_b_lds(sbias, f3_b, 100); __syncthreads();
  stage_lds<7, true, false>(gbuf, wbuf, sbias, hbuf, 25, lane);
  __syncthreads();
  // f4 + sigmoid -> out (input hbuf)
  load_w_lds(wbuf, f4_w, 100, 1); load_b_lds(sbias, f4_b, 1); __syncthreads();
  {
    v8f c = bias_tile(sbias, 0, lane);
    for (int kk = 0; kk < 25; ++kk) {
      const int k = 4 * kk + k2;
      v2f a = *(const v2f*)(hbuf + n * AST + k);
      v2f b = *(const v2f*)(wbuf + (k >> 1) * WPS + n * 2);
      c = wmma4(a, b, c);
    }
    if (n == 0) {   // lanes 0 and 16 hold column 0 (rows 0-7 / 8-15)
#pragma unroll
      for (int r = 0; r < 8; ++r) {
        const float v = c[r];
        out[itembase + 8 * hi + r] = 1.0f / (1.0f + __expf(-v));
      }
    }
  }
}

extern "C" void kernel_launch(void* const* d_in, const int* in_sizes, int n_in,
                              void* d_out, int out_size, void* d_ws, size_t ws_size,
                              hipStream_t stream) {
  const float* feat   = (const float*)d_in[0];
  const float* adj    = (const float*)d_in[1];
  const float* hw_emb = (const float*)d_in[2];
  const float* gc1_w  = (const float*)d_in[3];
  const float* gc1_b  = (const float*)d_in[4];
  const float* gc2_w  = (const float*)d_in[5];
  const float* gc2_b  = (const float*)d_in[6];
  const float* gc3_w  = (const float*)d_in[7];
  const float* gc3_b  = (const float*)d_in[8];
  const float* gc4_w  = (const float*)d_in[9];
  const float* gc4_b  = (const float*)d_in[10];
  const float* fc3_w  = (const float*)d_in[11];
  const float* fc3_b  = (const float*)d_in[12];
  const float* fc4_w  = (const float*)d_in[13];
  const float* fc4_b  = (const float*)d_in[14];
  const float* fc5_w  = (const float*)d_in[15];
  const float* fc5_b  = (const float*)d_in[16];
  const float* f0_w   = (const float*)d_in[17];
  const float* f0_b   = (const float*)d_in[18];
  const float* f1_w   = (const float*)d_in[19];
  const float* f1_b   = (const float*)d_in[20];
  const float* f2_w   = (const float*)d_in[21];
  const float* f2_b   = (const float*)d_in[22];
  const float* f3_w   = (const float*)d_in[23];
  const float* f3_b   = (const float*)d_in[24];
  const float* f4_w   = (const float*)d_in[25];
  const float* f4_b   = (const float*)d_in[26];
  const float* e0_w   = (const float*)d_in[27];
  const float* e0_b   = (const float*)d_in[28];
  const float* e1_w   = (const float*)d_in[29];
  const float* e1_b   = (const float*)d_in[30];
  float* out = (float*)d_out;

  const int Bc = in_sizes[0] / 64;              // feat is B*8*8
  const int grid = Bc / ITEMS_PER_BLOCK;        // 131072/128 = 1024

  (void)hipFuncSetAttribute(reinterpret_cast<const void*>(predictor_kernel),
                            hipFuncAttributeMaxDynamicSharedMemorySize, (int)LDS_BYTES);

  predictor_kernel<<<grid, BLOCK, LDS_BYTES, stream>>>(
      feat, adj, hw_emb,
      gc1_w, gc1_b, gc2_w, gc2_b, gc3_w, gc3_b, gc4_w, gc4_b,
      fc3_w, fc3_b, fc4_w, fc4_b, fc5_w, fc5_b,
      f0_w, f0_b, f1_w, f1_b, f2_w, f2_b, f3_w, f3_b, f4_w, f4_b,
      e0_w, e0_b, e1_w, e1_b, out);
}